// CantorExpert_58677843198308
// MI455X (gfx1250) — compile-verified
//
#include <hip/hip_runtime.h>
#include <math.h>

#define FULL      4096
#define EDIM      1024
#define SLICE     512
#define H1        128
#define SL_START  1024
#define NB        4
#define P_TOK     4096
#define FP_MIN_C  0.1875f
#define FP_MAX_C  0.4375f

typedef float v2f __attribute__((ext_vector_type(2)));
typedef float v8f __attribute__((ext_vector_type(8)));

// ---------------------------------------------------------------------------
// Kernel A: wave32 stream compaction of fingerprint mask -> sorted idx list.
// ---------------------------------------------------------------------------
__global__ void ck_compact(const float* __restrict__ fp, int* __restrict__ idx) {
  const int lane = threadIdx.x;              // launched with exactly 32 threads
  int base = 0;
  for (int i0 = 0; i0 < P_TOK; i0 += 32) {
    const int i = i0 + lane;
    const float f = fp[i];
    const bool m = (f >= FP_MIN_C) && (f < FP_MAX_C);
    const unsigned bal = __builtin_amdgcn_ballot_w32(m);
    const unsigned pre = __popc(bal & ((1u << lane) - 1u));
    if (m) idx[base + pre] = i;
    base += __popc(bal);
  }
}

// ---------------------------------------------------------------------------
// Kernel B: gather slice, gate MLP (exact GELU), write gated rows to my_g.
// One 128-thread block per (token j, batch b).
// ---------------------------------------------------------------------------
__global__ void ck_gate(const float* __restrict__ tokens, const int* __restrict__ idx,
                        const float* __restrict__ Wg1, const float* __restrict__ bg1,
                        const float* __restrict__ Wg2, const float* __restrict__ bg2,
                        const float* __restrict__ alpha,
                        float* __restrict__ myg, int N) {
  __shared__ float sx[SLICE];
  __shared__ float sp[H1];
  __shared__ float sg;
  const int j = blockIdx.x;
  const int b = blockIdx.y;
  const int t = threadIdx.x;                 // 128 threads
  const int p = idx[j];

  const float* src = tokens + ((size_t)b * P_TOK + (size_t)p) * FULL + SL_START;
  for (int k = t; k < SLICE; k += H1) sx[k] = src[k];
  __syncthreads();

  // hidden unit t: dot(my_row, W_g1[t,:]) + b, exact GELU
  const float* wr = Wg1 + (size_t)t * SLICE;
  float acc = bg1[t];
#pragma unroll 4
  for (int k = 0; k < SLICE; ++k) acc = fmaf(sx[k], wr[k], acc);
  const float h = 0.5f * acc * (1.0f + erff(acc * 0.70710678118654752f));
  sp[t] = h * Wg2[t];
  __syncthreads();

  for (int s = H1 / 2; s > 0; s >>= 1) {
    if (t < s) sp[t] += sp[t + s];
    __syncthreads();
  }
  if (t == 0) {
    const float gate = 1.0f / (1.0f + expf(-(sp[0] + bg2[0])));
    const float aw   = 1.0f / (1.0f + expf(-alpha[0]));
    sg = gate * aw + (1.0f - aw);
  }
  __syncthreads();

  const float g = sg;
  float* dst = myg + ((size_t)b * N + j) * SLICE;
  for (int k = t; k < SLICE; k += H1) dst[k] = sx[k] * g;
}

// ---------------------------------------------------------------------------
// Kernel C: Q/K/V GEMM in full fp32 via V_WMMA_F32_16X16X4_F32.
// OUT[m, n] = sum_k my_g[m, k] * W[n, k];  32x32 tile per wave (4 accumulators).
// f32 WMMA layouts:
//   A 16x4 : lanes 0-15 hold row M=l16, K = k,k+1 ; lanes 16-31 hold K = k+2,k+3
//   B 4x16 : lanes 0-15 hold col N=l16, K = k,k+1 ; lanes 16-31 hold K = k+2,k+3
//   C/D    : VGPR v holds row m0 + v (+8 for lanes 16-31), col = l16
// Out-of-range M rows: A row index is CLAMPED (always loads valid memory);
// garbage flows only into accumulator rows that the guarded store discards.
// This keeps the inner loop branch-free (no EXEC save/restore per load).
// ---------------------------------------------------------------------------
__device__ __forceinline__ v2f ld2(const float* p) { return *(const v2f*)p; }

__global__ void ck_qkv(const float* __restrict__ myg,
                       const float* __restrict__ Wq, const float* __restrict__ Wk,
                       const float* __restrict__ Wv,
                       float* __restrict__ out, int M) {
  const float* W = (blockIdx.z == 0) ? Wq : (blockIdx.z == 1 ? Wk : Wv);
  float* O = out + (size_t)blockIdx.z * (size_t)M * EDIM;

  const int tid  = threadIdx.x;              // 256 threads = 8 waves
  const int wave = tid >> 5;
  const int lane = tid & 31;
  const int l16  = lane & 15;
  const bool lo  = lane < 16;

  const int m0 = (blockIdx.y * 8 + wave) * 32;
  const int n0 = blockIdx.x * 32;
  if (m0 >= M) return;                       // uniform per wave; EXEC full below

  v8f c00 = {}, c01 = {}, c10 = {}, c11 = {};

  const int ra0 = m0 + l16;
  const int ra1 = m0 + 16 + l16;
  const int ca0 = (ra0 < M) ? ra0 : (M - 1); // clamp: always-valid, branch-free
  const int ca1 = (ra1 < M) ? ra1 : (M - 1);
  const int koff = lo ? 0 : 2;
  const float* a0p = myg + (size_t)ca0 * SLICE + koff;
  const float* a1p = myg + (size_t)ca1 * SLICE + koff;
  const float* b0p = W + (size_t)(n0 + l16) * SLICE + koff;
  const float* b1p = W + (size_t)(n0 + 16 + l16) * SLICE + koff;

#pragma unroll 8
  for (int k = 0; k < SLICE; k += 4) {
    v2f a0 = ld2(a0p + k);
    v2f a1 = ld2(a1p + k);
    v2f b0 = ld2(b0p + k);
    v2f b1 = ld2(b1p + k);
    c00 = __builtin_amdgcn_wmma_f32_16x16x4_f32(false, a0, false, b0, (short)0, c00, false, false);
    c01 = __builtin_amdgcn_wmma_f32_16x16x4_f32(false, a0, false, b1, (short)0, c01, false, false);
    c10 = __builtin_amdgcn_wmma_f32_16x16x4_f32(false, a1, false, b0, (short)0, c10, false, false);
    c11 = __builtin_amdgcn_wmma_f32_16x16x4_f32(false, a1, false, b1, (short)0, c11, false, false);
  }

  const int rb0 = m0 + (lo ? 0 : 8);
  const int rb1 = m0 + 16 + (lo ? 0 : 8);
#pragma unroll
  for (int v = 0; v < 8; ++v) {
    const int r0 = rb0 + v;
    const int r1 = rb1 + v;
    if (r0 < M) {
      O[(size_t)r0 * EDIM + n0 + l16]      = c00[v];
      O[(size_t)r0 * EDIM + n0 + 16 + l16] = c01[v];
    }
    if (r1 < M) {
      O[(size_t)r1 * EDIM + n0 + l16]      = c10[v];
      O[(size_t)r1 * EDIM + n0 + 16 + l16] = c11[v];
    }
  }
}

// ---------------------------------------------------------------------------
// Kernel D: pentachoron projections. One wave per output element:
//   out[which][v][m] = (Q_row . penta[v]) / ||penta[v]||
// ---------------------------------------------------------------------------
__global__ void ck_proj(const float* __restrict__ qkv, const float* __restrict__ penta,
                        float* __restrict__ outp, int M) {
  const int gw   = (int)((blockIdx.x * blockDim.x + threadIdx.x) >> 5);
  const int lane = threadIdx.x & 31;
  const int total = 3 * 5 * M;
  if (gw >= total) return;

  const int which = gw / (5 * M);
  const int rem   = gw % (5 * M);
  const int v     = rem / M;
  const int m     = rem % M;

  const float* q = qkv + ((size_t)which * M + m) * EDIM;
  const float* d = penta + (size_t)v * EDIM;

  float s1 = 0.0f, s2 = 0.0f;
  for (int k = lane; k < EDIM; k += 32) {
    const float dv = d[k];
    s1 = fmaf(q[k], dv, s1);
    s2 = fmaf(dv, dv, s2);
  }
#pragma unroll
  for (int off = 16; off > 0; off >>= 1) {
    s1 += __shfl_xor(s1, off, 32);
    s2 += __shfl_xor(s2, off, 32);
  }
  if (lane == 0)
    outp[(size_t)which * 5 * M + (size_t)v * M + m] = s1 / sqrtf(s2);
}

// ---------------------------------------------------------------------------
extern "C" void kernel_launch(void* const* d_in, const int* in_sizes, int n_in,
                              void* d_out, int out_size, void* d_ws, size_t ws_size,
                              hipStream_t stream) {
  const float* tokens = (const float*)d_in[0];
  const float* fps    = (const float*)d_in[1];
  const float* Wg1    = (const float*)d_in[2];
  const float* bg1    = (const float*)d_in[3];
  const float* Wg2    = (const float*)d_in[4];
  const float* bg2    = (const float*)d_in[5];
  const float* alpha  = (const float*)d_in[6];
  const float* Wq     = (const float*)d_in[7];
  const float* Wk     = (const float*)d_in[8];
  const float* Wv     = (const float*)d_in[9];
  const float* penta  = (const float*)d_in[10];
  float* out = (float*)d_out;

  // out_size = B*N*(3*EDIM) + B*N*15  ->  N recovered deterministically
  const int N = out_size / (NB * (3 * EDIM + 15));
  const int M = NB * N;

  int*   idx = (int*)d_ws;
  float* myg = (float*)((char*)d_ws + P_TOK * sizeof(int));

  ck_compact<<<1, 32, 0, stream>>>(fps, idx);
  if (N <= 0) return;

  ck_gate<<<dim3((unsigned)N, NB), H1, 0, stream>>>(tokens, idx, Wg1, bg1, Wg2, bg2,
                                                    alpha, myg, N);

  ck_qkv<<<dim3(EDIM / 32, (unsigned)((M + 255) / 256), 3), 256, 0, stream>>>(
      myg, Wq, Wk, Wv, out, M);

  const int totalw = 3 * 5 * M;
  ck_proj<<<(unsigned)((totalw + 7) / 8), 256, 0, stream>>>(
      out, penta, out + (size_t)3 * M * EDIM, M);
}